// ABCNN3_42417097017126
// MI455X (gfx1250) — compile-verified
//
#include <hip/hip_runtime.h>
#include <math.h>
#include <stdint.h>

typedef __attribute__((ext_vector_type(2))) float v2f;
typedef __attribute__((ext_vector_type(8))) float v8f;

#define B_    128
#define S_    256
#define D_    512
#define L_    2
#define LIN_  1024
#define FEAT_ 3072   // D*(1+L)*2

// ---------------------------------------------------------------------------
// CDNA5 async memory -> LDS helpers (tracked by ASYNCcnt)
// ---------------------------------------------------------------------------
__device__ __forceinline__ void async_ld_b32(float* lds, const float* g)
{
    uint32_t lo = (uint32_t)(uintptr_t)lds;   // low 32 bits of flat LDS addr = LDS offset
    asm volatile("global_load_async_to_lds_b32 %0, %1, off"
                 :: "v"(lo), "v"(g) : "memory");
}
__device__ __forceinline__ void async_ld_b128(float* lds, const float* g)
{
    uint32_t lo = (uint32_t)(uintptr_t)lds;
    asm volatile("global_load_async_to_lds_b128 %0, %1, off"
                 :: "v"(lo), "v"(g) : "memory");
}
__device__ __forceinline__ void wait_async0()
{
    asm volatile("s_wait_asynccnt 0x0" ::: "memory");
}

// ---------------------------------------------------------------------------
// Generic batched WMMA fp32 GEMM:  C[b](M,N) = A[b](M,K) * B[b](K,N)
// Operand transposes expressed via (row-stride, col-stride).
// SCORE=true fuses the match_score epilogue; otherwise optional bias add.
// Block: 128 threads (4 wave32s), 64x64 block tile, 32x32 per wave (2x2 frags).
// K staged in 16-wide chunks via double-buffered async loads to LDS.
// ---------------------------------------------------------------------------
template <bool SCORE>
__global__ __launch_bounds__(128) void wmma_gemm(
    const float* __restrict__ Ag, int rsA, int csA, long bsA,
    const float* __restrict__ Bg, int rsB, int csB, long bsB,
    float* __restrict__ Cg, long bsC,
    const float* __restrict__ bias,
    const float* __restrict__ nn1, const float* __restrict__ nn2,
    const float* __restrict__ mm1, const float* __restrict__ mm2,
    int M, int N, int K, int Sdim)
{
    // pitches: 20 floats (80B) and 68 floats (272B) are 16B multiples so b128
    // async transfers stay aligned; 20,68 mod 64 banks keep column reads
    // across 16 rows conflict-free.
    __shared__ __align__(16) float lA[2][64][20];   // [buf][m][k]
    __shared__ __align__(16) float lB[2][16][68];   // [buf][k][n]

    const int b = blockIdx.z;
    const float* Ab = Ag + (size_t)b * bsA;
    const float* Bb = Bg + (size_t)b * bsB;
    float*       Cb = Cg + (size_t)b * bsC;

    const int tid  = threadIdx.x;
    const int lane = tid & 31;
    const int wv   = tid >> 5;
    const int i0   = blockIdx.x * 64;
    const int j0   = blockIdx.y * 64;
    const int wr   = (wv >> 1) * 32;
    const int wc   = (wv & 1) * 32;
    const int l15  = lane & 15;
    const int khalf = (lane >> 4) << 1;   // 0 or 2

    v8f c00 = {}, c01 = {}, c10 = {}, c11 = {};

    auto stageA = [&](int k0, int bf) {
        if (csA == 1) {
            for (int idx = tid; idx < 256; idx += 128) {       // 256 x 16B chunks
                int m = idx >> 2, k4 = (idx & 3) << 2;
                async_ld_b128(&lA[bf][m][k4],
                              Ab + (size_t)(i0 + m) * rsA + (size_t)(k0 + k4));
            }
        } else {                                               // A^T: m contiguous
            for (int idx = tid; idx < 1024; idx += 128) {
                int m = idx & 63, kk = idx >> 6;
                async_ld_b32(&lA[bf][m][kk],
                             Ab + (size_t)(i0 + m) * rsA + (size_t)(k0 + kk) * csA);
            }
        }
    };
    auto stageB = [&](int k0, int bf) {
        if (csB == 1) {
            for (int idx = tid; idx < 256; idx += 128) {
                int kk = idx >> 4, j4 = (idx & 15) << 2;
                async_ld_b128(&lB[bf][kk][j4],
                              Bb + (size_t)(k0 + kk) * rsB + (size_t)(j0 + j4));
            }
        } else {                                               // B^T: k contiguous
            for (int idx = tid; idx < 1024; idx += 128) {
                int kk = idx & 15, j = idx >> 4;
                async_ld_b32(&lB[bf][kk][j],
                             Bb + (size_t)(k0 + kk) * rsB + (size_t)(j0 + j) * csB);
            }
        }
    };

    stageA(0, 0);
    stageB(0, 0);

    int bf = 0;
    for (int k0 = 0; k0 < K; k0 += 16, bf ^= 1) {
        wait_async0();        // my async writes to lX[bf] have landed in LDS
        __syncthreads();      // everyone's landed; prior chunk's readers done
        if (k0 + 16 < K) {    // prefetch next chunk into the other buffer
            stageA(k0 + 16, bf ^ 1);
            stageB(k0 + 16, bf ^ 1);
        }

        #pragma unroll
        for (int kk = 0; kk < 16; kk += 4) {
            const int ko = kk + khalf;
            v2f a0, a1, b0, b1;
            a0.x = lA[bf][wr + l15][ko];           a0.y = lA[bf][wr + l15][ko + 1];
            a1.x = lA[bf][wr + 16 + l15][ko];      a1.y = lA[bf][wr + 16 + l15][ko + 1];
            b0.x = lB[bf][ko][wc + l15];           b0.y = lB[bf][ko + 1][wc + l15];
            b1.x = lB[bf][ko][wc + 16 + l15];      b1.y = lB[bf][ko + 1][wc + 16 + l15];
            c00 = __builtin_amdgcn_wmma_f32_16x16x4_f32(false, a0, false, b0, (short)0, c00, false, false);
            c01 = __builtin_amdgcn_wmma_f32_16x16x4_f32(false, a0, false, b1, (short)0, c01, false, false);
            c10 = __builtin_amdgcn_wmma_f32_16x16x4_f32(false, a1, false, b0, (short)0, c10, false, false);
            c11 = __builtin_amdgcn_wmma_f32_16x16x4_f32(false, a1, false, b1, (short)0, c11, false, false);
        }
    }

    // ---- epilogue ----
    const int rbase = i0 + wr + ((lane >> 4) << 3);  // + v -> global row
    const int col0  = j0 + wc + l15;

    auto store = [&](v8f cc, int r0, int col) {
        float m2v = 0.f, n2v = 0.f, bv = 0.f;
        if (SCORE) {
            m2v = mm2[(size_t)b * Sdim + col];
            n2v = nn2[(size_t)b * Sdim + col];
        } else if (bias) {
            bv = bias[col];
        }
        #pragma unroll
        for (int v = 0; v < 8; ++v) {
            int r = r0 + v;
            float val = cc[v];
            if (SCORE) {
                float m1v = mm1[(size_t)b * Sdim + r];
                float n1v = nn1[(size_t)b * Sdim + r];
                float t = n1v + n2v - 2.f * m1v * m2v * val;
                t = fmaxf(t, 0.f);
                val = 1.f / (1.f + sqrtf(t));
            } else {
                val += bv;
            }
            Cb[(size_t)r * N + col] = val;
        }
    };
    store(c00, rbase,      col0);
    store(c01, rbase,      col0 + 16);
    store(c10, rbase + 16, col0);
    store(c11, rbase + 16, col0 + 16);
}

// ---------------------------------------------------------------------------
// Support kernels
// ---------------------------------------------------------------------------
__global__ void gather_embed(const int* __restrict__ tok, const float* __restrict__ emb,
                             float* __restrict__ out, int total)
{
    int idx = blockIdx.x * 256 + threadIdx.x;
    if (idx >= total) return;
    int d  = idx & (D_ - 1);
    int bs = idx >> 9;                  // D_ == 512
    out[idx] = emb[(size_t)tok[bs] * D_ + d];
}

__global__ void make_mask(const int* __restrict__ tok, float* __restrict__ m, int total)
{
    int i = blockIdx.x * 256 + threadIdx.x;
    if (i < total) m[i] = (tok[i] != 0) ? 1.f : 0.f;
}

__global__ void mean_over_s(const float* __restrict__ x, float* __restrict__ dst, int off)
{
    int d = blockIdx.x * 256 + threadIdx.x;   // gridDim.x = D/256
    int b = blockIdx.y;
    const float* p = x + (size_t)b * S_ * D_ + d;
    float acc = 0.f;
    for (int s = 0; s < S_; ++s) acc += p[(size_t)s * D_];
    dst[(size_t)b * FEAT_ + off + d] = acc * (1.f / S_);
}

__global__ __launch_bounds__(256) void row_norm(const float* __restrict__ x,
                                                const float* __restrict__ m,
                                                float* __restrict__ n)
{
    __shared__ float red[256];
    int row = blockIdx.x;                       // b*S + s
    const float* p = x + (size_t)row * D_;
    float acc = 0.f;
    for (int d = threadIdx.x; d < D_; d += 256) { float v = p[d]; acc += v * v; }
    red[threadIdx.x] = acc; __syncthreads();
    for (int st = 128; st > 0; st >>= 1) {
        if (threadIdx.x < st) red[threadIdx.x] += red[threadIdx.x + st];
        __syncthreads();
    }
    if (threadIdx.x == 0) n[row] = m[row] * red[0];
}

__global__ void conv_tanh(const float* __restrict__ base, const float* __restrict__ feat,
                          const float* __restrict__ kw, const float* __restrict__ kb,
                          float* __restrict__ out, int total)
{
    int idx = blockIdx.x * 256 + threadIdx.x;
    if (idx >= total) return;
    int d  = idx & (D_ - 1);
    int bs = idx >> 9;
    int s  = bs & (S_ - 1);
    float acc = kb[0];
    #pragma unroll
    for (int dh = -1; dh <= 1; ++dh) {
        int ss = s + dh;
        if (ss < 0 || ss >= S_) continue;
        #pragma unroll
        for (int dw = -1; dw <= 1; ++dw) {
            int dd = d + dw;
            if (dd < 0 || dd >= D_) continue;
            int w = (dh + 1) * 3 + (dw + 1);
            size_t off = (size_t)(bs + dh) * D_ + dd;
            acc += base[off] * kw[w] + feat[off] * kw[9 + w];
        }
    }
    out[idx] = tanhf(acc);
}

__global__ __launch_bounds__(256) void row_sum(const float* __restrict__ A, float* __restrict__ w)
{
    __shared__ float red[256];
    int row = blockIdx.x;                       // b*S + i ; S_ == blockDim
    red[threadIdx.x] = A[(size_t)row * S_ + threadIdx.x];
    __syncthreads();
    for (int st = 128; st > 0; st >>= 1) {
        if (threadIdx.x < st) red[threadIdx.x] += red[threadIdx.x + st];
        __syncthreads();
    }
    if (threadIdx.x == 0) w[row] = red[0];
}

__global__ void col_sum(const float* __restrict__ A, float* __restrict__ w)
{
    int b = blockIdx.x;
    int j = threadIdx.x;                        // 256 threads
    const float* p = A + (size_t)b * S_ * S_ + j;
    float acc = 0.f;
    for (int i = 0; i < S_; ++i) acc += p[(size_t)i * S_];
    w[b * S_ + j] = acc;
}

__global__ void pool_residual(const float* __restrict__ o, const float* __restrict__ w,
                              float* __restrict__ s, int total)
{
    int idx = blockIdx.x * 256 + threadIdx.x;
    if (idx >= total) return;
    int bs = idx >> 9;
    int ss = bs & (S_ - 1);
    float acc = o[idx] * w[bs];
    if (ss > 0)        acc += o[idx - D_] * w[bs - 1];
    if (ss < S_ - 1)   acc += o[idx + D_] * w[bs + 1];
    s[idx] += acc * (1.f / 3.f);
}

__global__ __launch_bounds__(256) void ln_relu(const float* __restrict__ y,
                                               const float* __restrict__ g,
                                               const float* __restrict__ bb,
                                               float* __restrict__ h)
{
    __shared__ float rs[256], rq[256];
    int b = blockIdx.x;
    const float* p = y + (size_t)b * LIN_;
    float v[4], s0 = 0.f, s1 = 0.f;
    #pragma unroll
    for (int q = 0; q < 4; ++q) {
        v[q] = p[threadIdx.x + q * 256];
        s0 += v[q]; s1 += v[q] * v[q];
    }
    rs[threadIdx.x] = s0; rq[threadIdx.x] = s1; __syncthreads();
    for (int st = 128; st > 0; st >>= 1) {
        if (threadIdx.x < st) { rs[threadIdx.x] += rs[threadIdx.x + st];
                                rq[threadIdx.x] += rq[threadIdx.x + st]; }
        __syncthreads();
    }
    float mu  = rs[0] * (1.f / LIN_);
    float var = rq[0] * (1.f / LIN_) - mu * mu;
    float inv = rsqrtf(var + 1e-5f);
    #pragma unroll
    for (int q = 0; q < 4; ++q) {
        int c = threadIdx.x + q * 256;
        float t = (v[q] - mu) * inv * g[c] + bb[c];
        h[(size_t)b * LIN_ + c] = fmaxf(t, 0.f);
    }
}

__global__ __launch_bounds__(256) void fc2_softmax(const float* __restrict__ h,
                                                   const float* __restrict__ w2,
                                                   const float* __restrict__ b2,
                                                   float* __restrict__ out)
{
    __shared__ float r0[256], r1[256];
    int b = blockIdx.x;
    const float* p = h + (size_t)b * LIN_;
    float a0 = 0.f, a1 = 0.f;
    for (int k = threadIdx.x; k < LIN_; k += 256) {
        float hv = p[k];
        a0 += hv * w2[k * 2];
        a1 += hv * w2[k * 2 + 1];
    }
    r0[threadIdx.x] = a0; r1[threadIdx.x] = a1; __syncthreads();
    for (int st = 128; st > 0; st >>= 1) {
        if (threadIdx.x < st) { r0[threadIdx.x] += r0[threadIdx.x + st];
                                r1[threadIdx.x] += r1[threadIdx.x + st]; }
        __syncthreads();
    }
    if (threadIdx.x == 0) {
        float l0 = r0[0] + b2[0], l1 = r1[0] + b2[1];
        float mx = fmaxf(l0, l1);
        float e0 = expf(l0 - mx), e1 = expf(l1 - mx);
        float inv = 1.f / (e0 + e1);
        out[b * 2]     = e0 * inv;
        out[b * 2 + 1] = e1 * inv;
    }
}

// ---------------------------------------------------------------------------
extern "C" void kernel_launch(void* const* d_in, const int* in_sizes, int n_in,
                              void* d_out, int out_size, void* d_ws, size_t ws_size,
                              hipStream_t stream)
{
    (void)in_sizes; (void)n_in; (void)out_size; (void)ws_size;
    const int*   s1_tok = (const int*)d_in[0];
    const int*   s2_tok = (const int*)d_in[1];
    const float* embeds = (const float*)d_in[2];
    const float* Ws     = (const float*)d_in[3];
    const float* conv_k = (const float*)d_in[4];
    const float* conv_b = (const float*)d_in[5];
    const float* fc_w1  = (const float*)d_in[6];
    const float* fc_b1  = (const float*)d_in[7];
    const float* ln_g   = (const float*)d_in[8];
    const float* ln_b   = (const float*)d_in[9];
    const float* fc_w2  = (const float*)d_in[10];
    const float* fc_b2  = (const float*)d_in[11];
    float* out = (float*)d_out;

    // scratch layout
    float* ws = (float*)d_ws;
    const size_t NSD = (size_t)B_ * S_ * D_;
    float* s1 = ws; ws += NSD;
    float* s2 = ws; ws += NSD;
    float* f1 = ws; ws += NSD;
    float* f2 = ws; ws += NSD;
    float* o1 = ws; ws += NSD;
    float* o2 = ws; ws += NSD;
    float* Am = ws; ws += (size_t)B_ * S_ * S_;
    float* m1 = ws; ws += B_ * S_;
    float* m2 = ws; ws += B_ * S_;
    float* n1 = ws; ws += B_ * S_;
    float* n2 = ws; ws += B_ * S_;
    float* w1 = ws; ws += B_ * S_;
    float* w2 = ws; ws += B_ * S_;
    float* xf = ws; ws += (size_t)B_ * FEAT_;
    float* y  = ws; ws += (size_t)B_ * LIN_;
    float* h  = ws; ws += (size_t)B_ * LIN_;

    const int totSD = (int)NSD;
    dim3 blk(256);
    dim3 gSD((totSD + 255) / 256);
    dim3 gBS((B_ * S_ + 255) / 256);

    gather_embed<<<gSD, blk, 0, stream>>>(s1_tok, embeds, s1, totSD);
    gather_embed<<<gSD, blk, 0, stream>>>(s2_tok, embeds, s2, totSD);
    make_mask<<<gBS, blk, 0, stream>>>(s1_tok, m1, B_ * S_);
    make_mask<<<gBS, blk, 0, stream>>>(s2_tok, m2, B_ * S_);
    mean_over_s<<<dim3(D_ / 256, B_), blk, 0, stream>>>(s1, xf, 0);
    mean_over_s<<<dim3(D_ / 256, B_), blk, 0, stream>>>(s2, xf, FEAT_ / 2);

    for (int i = 0; i < L_; ++i) {
        row_norm<<<dim3(B_ * S_), blk, 0, stream>>>(s1, m1, n1);
        row_norm<<<dim3(B_ * S_), blk, 0, stream>>>(s2, m2, n2);
        // A = match_score(s1, s2): cross = s1 @ s2^T fused with score transform
        wmma_gemm<true><<<dim3(S_ / 64, S_ / 64, B_), dim3(128), 0, stream>>>(
            s1, D_, 1, (long)S_ * D_,
            s2, 1, D_, (long)S_ * D_,
            Am, (long)S_ * S_,
            nullptr, n1, n2, m1, m2, S_, S_, D_, S_);
        const float* Wi = Ws + (size_t)i * S_ * D_;
        // f1 = A @ W
        wmma_gemm<false><<<dim3(S_ / 64, D_ / 64, B_), dim3(128), 0, stream>>>(
            Am, S_, 1, (long)S_ * S_,
            Wi, D_, 1, 0L,
            f1, (long)S_ * D_,
            nullptr, nullptr, nullptr, nullptr, nullptr, S_, D_, S_, 0);
        // f2 = A^T @ W
        wmma_gemm<false><<<dim3(S_ / 64, D_ / 64, B_), dim3(128), 0, stream>>>(
            Am, 1, S_, (long)S_ * S_,
            Wi, D_, 1, 0L,
            f2, (long)S_ * D_,
            nullptr, nullptr, nullptr, nullptr, nullptr, S_, D_, S_, 0);
        conv_tanh<<<gSD, blk, 0, stream>>>(s1, f1, conv_k + i * 18, conv_b + i, o1, totSD);
        conv_tanh<<<gSD, blk, 0, stream>>>(s2, f2, conv_k + i * 18, conv_b + i, o2, totSD);
        mean_over_s<<<dim3(D_ / 256, B_), blk, 0, stream>>>(o1, xf, (i + 1) * D_);
        mean_over_s<<<dim3(D_ / 256, B_), blk, 0, stream>>>(o2, xf, FEAT_ / 2 + (i + 1) * D_);
        // A2 = match_score(o1, o2)
        row_norm<<<dim3(B_ * S_), blk, 0, stream>>>(o1, m1, n1);
        row_norm<<<dim3(B_ * S_), blk, 0, stream>>>(o2, m2, n2);
        wmma_gemm<true><<<dim3(S_ / 64, S_ / 64, B_), dim3(128), 0, stream>>>(
            o1, D_, 1, (long)S_ * D_,
            o2, 1, D_, (long)S_ * D_,
            Am, (long)S_ * S_,
            nullptr, n1, n2, m1, m2, S_, S_, D_, S_);
        row_sum<<<dim3(B_ * S_), blk, 0, stream>>>(Am, w1);
        col_sum<<<dim3(B_), blk, 0, stream>>>(Am, w2);
        pool_residual<<<gSD, blk, 0, stream>>>(o1, w1, s1, totSD);
        pool_residual<<<gSD, blk, 0, stream>>>(o2, w2, s2, totSD);
    }

    // y = xf @ fc_w1 + fc_b1  (M=128, N=1024, K=3072)
    wmma_gemm<false><<<dim3(B_ / 64, LIN_ / 64, 1), dim3(128), 0, stream>>>(
        xf, FEAT_, 1, 0L,
        fc_w1, LIN_, 1, 0L,
        y, 0L,
        fc_b1, nullptr, nullptr, nullptr, nullptr, B_, LIN_, FEAT_, 0);
    ln_relu<<<dim3(B_), blk, 0, stream>>>(y, ln_g, ln_b, h);
    fc2_softmax<<<dim3(B_), blk, 0, stream>>>(h, fc_w2, fc_b2, out);
}